// HyperbolicLayer_76390288327369
// MI455X (gfx1250) — compile-verified
//
#include <hip/hip_runtime.h>
#include <math.h>

// HyperbolicLayer on Poincare ball, D=8192.
// Only W[:,0] is used -> whole problem is ~200 KFLOP + ~1 MB of traffic.
// Strategy: single workgroup (1024 threads = 32 wave32 waves), all vectors in
// registers, 5 block-level reduction rounds. The strided W-column gather
// (stride 32 KB -> 8192 cachelines, pure latency) is issued at kernel entry
// via CDNA5 async global->LDS loads (ASYNCcnt) and overlapped with phase-1.

#define DVEC 8192
#define BLOCK 1024
#define PER (DVEC / BLOCK) // 8 elements per thread

#if defined(__AMDGCN__) && __has_builtin(__builtin_amdgcn_global_load_async_to_lds_b32)
#define HAVE_ASYNC_LDS 1
#else
#define HAVE_ASYNC_LDS 0
#endif

typedef __attribute__((address_space(1))) int g_int;
typedef __attribute__((address_space(3))) int l_int;

__device__ __forceinline__ void wait_async0() {
#if __has_builtin(__builtin_amdgcn_s_wait_asynccnt)
  __builtin_amdgcn_s_wait_asynccnt(0);
#else
  asm volatile("s_wait_asynccnt 0" ::: "memory");
#endif
}

__device__ __forceinline__ float wave_sum(float v) {
  // wave32 shuffle-xor reduction tree
  v += __shfl_xor(v, 16, 32);
  v += __shfl_xor(v, 8, 32);
  v += __shfl_xor(v, 4, 32);
  v += __shfl_xor(v, 2, 32);
  v += __shfl_xor(v, 1, 32);
  return v;
}

// Reduce n (<=3) values across the 32-wave workgroup. red is LDS [n][32].
__device__ __forceinline__ void block_sum(float* vals, int n, float* red,
                                          int wave, int lane) {
  for (int k = 0; k < n; ++k) {
    float r = wave_sum(vals[k]);
    if (lane == 0) red[k * 32 + wave] = r;
  }
  __syncthreads();
  for (int k = 0; k < n; ++k) {
    float s = 0.0f;
#pragma unroll
    for (int w = 0; w < 32; ++w) s += red[k * 32 + w]; // broadcast reads
    vals[k] = s;
  }
  __syncthreads(); // red[] reused next round
}

__global__ __launch_bounds__(BLOCK) void hyperbolic_layer_kernel(
    const float* __restrict__ X, const float* __restrict__ W,
    const float* __restrict__ b, float* __restrict__ out) {
  __shared__ float lds_w[DVEC];   // W[:,0] staged by async DMA loads
  __shared__ float red[3 * 32];   // reduction partials

  const int t = threadIdx.x;
  const int wave = t >> 5;
  const int lane = t & 31;

  // ---- Kick off the strided W[:,0] gather ASAP (overlaps with phase 1) ----
#if HAVE_ASYNC_LDS
#pragma unroll
  for (int j = 0; j < PER; ++j) {
    const int i = j * BLOCK + t;
    __builtin_amdgcn_global_load_async_to_lds_b32(
        (g_int*)(W + (size_t)i * DVEC),
        (l_int*)(lds_w + i),
        /*imm offset*/ 0, /*cpol*/ 0);
  }
#else
  float wreg[PER];
#pragma unroll
  for (int j = 0; j < PER; ++j) {
    const int i = j * BLOCK + t;
    wreg[j] = W[(size_t)i * DVEC];
  }
#endif

  // ---- Load x and b (coalesced) ----
  float x[PER], bb[PER];
#pragma unroll
  for (int j = 0; j < PER; ++j) {
    const int i = j * BLOCK + t;
    x[j] = X[i];
    bb[j] = b[i];
  }

  // ================= Phase 1: logx = log_map(x) =================
  // p = 0.5 everywhere; un2 = 0.25*D = 2048; 2/pcf = 2*(1-2048) = -4094
  float vals[3];
  {
    float sx = 0.0f, sx2 = 0.0f;
#pragma unroll
    for (int j = 0; j < PER; ++j) { sx += x[j]; sx2 += x[j] * x[j]; }
    vals[0] = sx; vals[1] = sx2;
  }
  block_sum(vals, 2, red, wave, lane);
  const float sumx = vals[0], vn2 = vals[1];

  const float uv1 = -0.5f * sumx;
  const float un2_1 = 0.25f * (float)DVEC; // 2048
  const float cA = 1.0f + 2.0f * uv1 + vn2;
  const float cB = 1.0f - un2_1;           // -2047
  const float den1 = 1.0f + 2.0f * uv1 + un2_1 * vn2;
  const float inv_den1 = 1.0f / den1;

  float m[PER];
  {
    float sm2 = 0.0f;
#pragma unroll
    for (int j = 0; j < PER; ++j) {
      m[j] = (cA * (-0.5f) + cB * x[j]) * inv_den1;
      sm2 += m[j] * m[j];
    }
    vals[0] = sm2;
  }
  block_sum(vals, 1, red, wave, lane);
  const float mn = sqrtf(vals[0]);
  const float two_over_pcf = 2.0f * (1.0f - 0.25f * (float)DVEC); // -4094
  const float logc = two_over_pcf * atanhf(mn - 0.5f) / mn;

  // ================= Phase 2: u = W[:,0] * logx =================
  float u[PER];
#if HAVE_ASYNC_LDS
  wait_async0(); // each wave reads back only the LDS words its lanes fetched
#pragma unroll
  for (int j = 0; j < PER; ++j) u[j] = lds_w[j * BLOCK + t] * (logc * m[j]);
#else
#pragma unroll
  for (int j = 0; j < PER; ++j) u[j] = wreg[j] * (logc * m[j]);
#endif

  // ================= Phase 3: e = exponential_map(u) =================
  {
    float su2 = 0.0f;
#pragma unroll
    for (int j = 0; j < PER; ++j) su2 += u[j] * u[j];
    vals[0] = su2;
  }
  block_sum(vals, 1, red, wave, lane);
  const float un = sqrtf(vals[0]);
  const float pcf2 = 1.0f / (1.0f - 1e-10f * (float)DVEC);
  const float tscale = tanhf(0.5f * pcf2 * un) / un;

  float s[PER];
  {
    float ss = 0.0f, ss2 = 0.0f;
#pragma unroll
    for (int j = 0; j < PER; ++j) {
      s[j] = tscale * u[j];
      ss += s[j];
      ss2 += s[j] * s[j];
    }
    vals[0] = ss; vals[1] = ss2;
  }
  block_sum(vals, 2, red, wave, lane);
  const float uv2 = 1e-5f * vals[0];
  const float un2_2 = 1e-10f * (float)DVEC;
  const float vn2_2 = vals[1];
  const float cA2 = 1.0f + 2.0f * uv2 + vn2_2;
  const float cB2 = 1.0f - un2_2;
  const float inv_den2 = 1.0f / (1.0f + 2.0f * uv2 + un2_2 * vn2_2);

  float e[PER];
  {
    float deb = 0.0f, dee = 0.0f, dbbv = 0.0f;
#pragma unroll
    for (int j = 0; j < PER; ++j) {
      e[j] = (cA2 * 1e-5f + cB2 * s[j]) * inv_den2;
      deb += e[j] * bb[j];
      dee += e[j] * e[j];
      dbbv += bb[j] * bb[j];
    }
    vals[0] = deb; vals[1] = dee; vals[2] = dbbv;
  }
  // ================= Phase 4: z = mobius_add(e, b) =================
  block_sum(vals, 3, red, wave, lane);
  const float uv3 = vals[0], un2_3 = vals[1], vn2_3 = vals[2];
  const float cA3 = 1.0f + 2.0f * uv3 + vn2_3;
  const float cB3 = 1.0f - un2_3;
  const float inv_den3 = 1.0f / (1.0f + 2.0f * uv3 + un2_3 * vn2_3);

#pragma unroll
  for (int j = 0; j < PER; ++j) {
    const int i = j * BLOCK + t;
    out[i] = (cA3 * e[j] + cB3 * bb[j]) * inv_den3;
  }
}

extern "C" void kernel_launch(void* const* d_in, const int* in_sizes, int n_in,
                              void* d_out, int out_size, void* d_ws, size_t ws_size,
                              hipStream_t stream) {
  const float* X = (const float*)d_in[0]; // (1, 8192)
  const float* W = (const float*)d_in[1]; // (8192, 8192), only column 0 used
  const float* b = (const float*)d_in[2]; // (8192,)
  float* out = (float*)d_out;             // (1, 8192)
  (void)in_sizes; (void)n_in; (void)out_size; (void)d_ws; (void)ws_size;

  hipLaunchKernelGGL(hyperbolic_layer_kernel, dim3(1), dim3(BLOCK), 0, stream,
                     X, W, b, out);
}